// DifferentiableKLA_72885595013194
// MI455X (gfx1250) — compile-verified
//
#include <hip/hip_runtime.h>
#include <math.h>

#define WVL_BLOCK   256   // threads per block, one wavelength per thread
#define BATCH_CHUNK 64    // batches per block (thickness tile staged via TDM)

typedef unsigned int u32x4 __attribute__((ext_vector_type(4)));
typedef int          i32x4 __attribute__((ext_vector_type(4)));
typedef int          i32x8 __attribute__((ext_vector_type(8)));

__global__ __launch_bounds__(WVL_BLOCK)
void tmm_reflectance_kernel(const float*  __restrict__ d_phys,   // [B,7]
                            const float2* __restrict__ nSi,      // [W] complex64
                            const float2* __restrict__ nO2,      // [W]
                            const float2* __restrict__ nN4,      // [W]
                            const float*  __restrict__ lam,      // [W]
                            float*        __restrict__ out,      // [B,W]
                            int W, int B)
{
    __shared__ __align__(16) float sd[BATCH_CHUNK * 7];

    const int  w      = blockIdx.x * WVL_BLOCK + threadIdx.x;
    const int  b0     = blockIdx.y * BATCH_CHUNK;
    const int  rows   = min(BATCH_CHUNK, B - b0);
    const bool active = (w < W);
    const int  wc     = active ? w : (W - 1);   // clamp so inactive lanes load in-range

    // ---------------- Stage thickness tile into LDS via Tensor Data Mover ----------------
#if __has_builtin(__builtin_amdgcn_tensor_load_to_lds)
    if (threadIdx.x < 32) {        // one wave issues the DMA for the whole block
        unsigned long long ga  = (unsigned long long)(const void*)(d_phys + (size_t)b0 * 7);
        unsigned           lds = (unsigned)(size_t)(void*)sd;   // low 32 bits = LDS byte offset

        // D# group 0: count=1, lds_addr, global_addr[56:0], type=2
        u32x4 g0;
        g0.x = 1u;
        g0.y = lds;
        g0.z = (unsigned)(ga & 0xFFFFFFFFu);
        g0.w = (unsigned)((ga >> 32) & 0x01FFFFFFu) | (2u << 30);

        // D# group 1: 2-D tensor, data_size=4B, dim0=7 (stride 7), tile 7 x rows
        const unsigned td0 = 7u, td1 = (unsigned)(B - b0);
        const unsigned tl0 = 7u, tl1 = (unsigned)rows;
        i32x8 g1;
        g1[0] = (int)(2u << 16);                                          // data_size=2 -> 4 bytes
        g1[1] = (int)((td0 & 0xFFFFu) << 16);                             // tensor_dim0 lo16
        g1[2] = (int)(((td0 >> 16) & 0xFFFFu) | ((td1 & 0xFFFFu) << 16)); // td0 hi | td1 lo
        g1[3] = (int)(((td1 >> 16) & 0xFFFFu) | ((tl0 & 0xFFFFu) << 16)); // td1 hi | tile_dim0
        g1[4] = (int)(tl1 & 0xFFFFu);                                     // tile_dim1 | tile_dim2=0
        g1[5] = 7;                                                        // tensor_dim0_stride lo32
        g1[6] = 0;
        g1[7] = 0;

        i32x4 gz4  = {0, 0, 0, 0};               // groups 2/3 unused (<=2D tensor)
        i32x8 gz8  = {0, 0, 0, 0, 0, 0, 0, 0};   // trailing group (unused), zero-filled
        // 6-arg toolchain variant: (g0, g1, g2, g3, g4, cpol)
        __builtin_amdgcn_tensor_load_to_lds(g0, g1, gz4, gz4, gz8, 0);
    }
#else
    for (int i = threadIdx.x; i < rows * 7; i += WVL_BLOCK)
        sd[i] = d_phys[(size_t)b0 * 7 + i];
#endif

    // ---------------- Per-wavelength interface constants (overlaps the DMA) ----------------
    const float  lw = lam[wc];
    const float  k0 = 6.283185307179586f / lw;
    const float2 vO = nO2[wc];
    const float2 vN = nN4[wc];
    const float2 vS = nSi[wc];

    // Air | SiO2 | Si3N4 | SiO2 | Si3N4 | SiO2 | Si3N4 | SiO2 | Si
    float nre[9], nim[9];
    nre[0] = 1.f;  nim[0] = 0.f;
    nre[1] = vO.x; nim[1] = vO.y;
    nre[2] = vN.x; nim[2] = vN.y;
    nre[3] = vO.x; nim[3] = vO.y;
    nre[4] = vN.x; nim[4] = vN.y;
    nre[5] = vO.x; nim[5] = vO.y;
    nre[6] = vN.x; nim[6] = vN.y;
    nre[7] = vO.x; nim[7] = vO.y;
    nre[8] = vS.x; nim[8] = vS.y;

    float rr[8], ri[8], tr[8], ti[8];   // Fresnel r and 1/t per interface
    float cr[7], ci[7];                 // k0 * n_next for interior layers
#pragma unroll
    for (int i = 0; i < 8; ++i) {
        const float sr = nre[i] + nre[i + 1], si = nim[i] + nim[i + 1];
        const float br = nre[i] - nre[i + 1], bi = nim[i] - nim[i + 1];
        const float inv1 = 1.0f / (sr * sr + si * si);
        rr[i] = (br * sr + bi * si) * inv1;            // r = (Nc-Nn)/(Nc+Nn)
        ri[i] = (bi * sr - br * si) * inv1;
        const float dr2 = 2.f * nre[i], di2 = 2.f * nim[i];
        const float inv2 = 1.0f / (dr2 * dr2 + di2 * di2);
        tr[i] = (sr * dr2 + si * di2) * inv2;          // 1/t = (Nc+Nn)/(2 Nc)
        ti[i] = (si * dr2 - sr * di2) * inv2;
        if (i < 7) { cr[i] = k0 * nre[i + 1]; ci[i] = k0 * nim[i + 1]; }
    }

#if __has_builtin(__builtin_amdgcn_tensor_load_to_lds)
    if (threadIdx.x < 32) __builtin_amdgcn_s_wait_tensorcnt(0);
    asm volatile("" :: "r"((float*)sd) : "memory");  // TDM wrote LDS behind the compiler's back
#endif
    __syncthreads();

    // ---------------- Batch loop: register-resident complex 2x2 recurrence ----------------
    for (int bl = 0; bl < rows; ++bl) {
        float dth[7];
#pragma unroll
        for (int j = 0; j < 7; ++j) dth[j] = sd[bl * 7 + j];   // LDS broadcast

        float M00r = 1.f, M00i = 0.f, M01r = 0.f, M01i = 0.f;
        float M10r = 0.f, M10i = 0.f, M11r = 1.f, M11i = 0.f;

#pragma unroll
        for (int i = 0; i < 8; ++i) {
            // m = (Mx + My*r) * inv_t   (complex)
            const float ar = M00r + (M01r * rr[i] - M01i * ri[i]);
            const float ai = M00i + (M01r * ri[i] + M01i * rr[i]);
            const float br = (M00r * rr[i] - M00i * ri[i]) + M01r;
            const float bi = (M00r * ri[i] + M00i * rr[i]) + M01i;
            const float gr = M10r + (M11r * rr[i] - M11i * ri[i]);
            const float gi = M10i + (M11r * ri[i] + M11i * rr[i]);
            const float hr = (M10r * rr[i] - M10i * ri[i]) + M11r;
            const float hi = (M10r * ri[i] + M10i * rr[i]) + M11i;
            M00r = ar * tr[i] - ai * ti[i];  M00i = ar * ti[i] + ai * tr[i];
            M01r = br * tr[i] - bi * ti[i];  M01i = br * ti[i] + bi * tr[i];
            M10r = gr * tr[i] - gi * ti[i];  M10i = gr * ti[i] + gi * tr[i];
            M11r = hr * tr[i] - hi * ti[i];  M11i = hr * ti[i] + hi * tr[i];

            if (i < 7) {
                // phi = k0 * n * d (complex); P = e^{-i phi}, Pi = e^{+i phi}
                const float phr = cr[i] * dth[i];
                const float phi = ci[i] * dth[i];
                float s, c;
                __sincosf(phr, &s, &c);
                const float ep = __expf(phi), em = __expf(-phi);
                const float Pr = ep * c,  Pi_ = -ep * s;   // e^{-i phi}
                const float Qr = em * c,  Qi  =  em * s;   // e^{+i phi}
                float t;
                t = M00r * Pr - M00i * Pi_; M00i = M00r * Pi_ + M00i * Pr; M00r = t;
                t = M01r * Qr - M01i * Qi;  M01i = M01r * Qi  + M01i * Qr; M01r = t;
                t = M10r * Pr - M10i * Pi_; M10i = M10r * Pi_ + M10i * Pr; M10r = t;
                t = M11r * Qr - M11i * Qi;  M11i = M11r * Qi  + M11i * Qr; M11r = t;
            }
        }

        const float num = M10r * M10r + M10i * M10i;
        const float den = M00r * M00r + M00i * M00i;
        if (active)
            out[(size_t)(b0 + bl) * W + w] = num / den;   // R = |M10/M00|^2
    }
}

extern "C" void kernel_launch(void* const* d_in, const int* in_sizes, int n_in,
                              void* d_out, int out_size, void* d_ws, size_t ws_size,
                              hipStream_t stream) {
    const float*  d_phys = (const float*)d_in[0];
    const float2* nSi    = (const float2*)d_in[1];
    const float2* nO2    = (const float2*)d_in[2];
    const float2* nN4    = (const float2*)d_in[3];
    const float*  lam    = (const float*)d_in[4];
    float* out = (float*)d_out;

    const int W = in_sizes[4];        // 801
    const int B = in_sizes[0] / 7;    // 16384

    dim3 block(WVL_BLOCK);
    dim3 grid((W + WVL_BLOCK - 1) / WVL_BLOCK,
              (B + BATCH_CHUNK - 1) / BATCH_CHUNK);
    hipLaunchKernelGGL(tmm_reflectance_kernel, grid, block, 0, stream,
                       d_phys, nSi, nO2, nN4, lam, out, W, B);
}